// BasicRelationModule_9663676416637
// MI455X (gfx1250) — compile-verified
//
#include <hip/hip_runtime.h>
#include <hip/hip_bf16.h>
#include <math.h>

typedef float v2f __attribute__((ext_vector_type(2)));
typedef float v8f __attribute__((ext_vector_type(8)));

#define BATCH 2
#define NPTS  10000
#define LDIM  128
#define HDIM  64
#define KSEL  32
#define TILES (NPTS / 16)   // 625
#define EPSV  1e-8f

// ---------------- Kernel 0: row-normalize x -> xn ----------------
__global__ void normalize_rows(const float* __restrict__ x, float* __restrict__ xn) {
  int row  = blockIdx.x * 8 + (threadIdx.x >> 5);   // one wave32 per row
  int lane = threadIdx.x & 31;
  if (row >= BATCH * NPTS) return;
  const float4* xin = (const float4*)(x + (size_t)row * LDIM);
  float4 v = xin[lane];
  float s = v.x * v.x + v.y * v.y + v.z * v.z + v.w * v.w;
  for (int off = 16; off > 0; off >>= 1) s += __shfl_xor(s, off, 32);
  float scale = 1.0f / (sqrtf(s) + EPSV);
  float4 o; o.x = v.x * scale; o.y = v.y * scale; o.z = v.z * scale; o.w = v.w * scale;
  ((float4*)(xn + (size_t)row * LDIM))[lane] = o;
}

// ---------------- Kernel 1: h = x @ W + b ----------------
__global__ void project_h(const float* __restrict__ x, const float* __restrict__ W,
                          const float* __restrict__ b, float* __restrict__ h) {
  __shared__ float Ws[LDIM * HDIM];   // 32 KB
  __shared__ float Xs[4 * LDIM];      // 2 KB
  int tid = threadIdx.x;
  for (int i = tid; i < LDIM * HDIM; i += 256) Ws[i] = W[i];
  int row0 = blockIdx.x * 4;
  for (int i = tid; i < 4 * LDIM; i += 256) Xs[i] = x[(size_t)row0 * LDIM + i];
  __syncthreads();
  int r = tid >> 6;
  int c = tid & 63;
  float acc = b[c];
  #pragma unroll 8
  for (int l = 0; l < LDIM; ++l) acc += Xs[r * LDIM + l] * Ws[l * HDIM + c];
  h[(size_t)(row0 + r) * HDIM + c] = acc;
}

// ---------------- Kernel 2: sim (WMMA f32) + streaming top-k + output ----------------
__global__ __launch_bounds__(128, 1)
void sim_topk_out(const float* __restrict__ xn, const float* __restrict__ h,
                  float* __restrict__ out) {
  __shared__ float As[16 * LDIM];        // 8 KB  (reused as merged lists later)
  __shared__ float Bs[4 * 16 * LDIM];    // 32 KB (per-wave B tile; prefix reused as D tile)
  __shared__ float topv[4 * 16 * KSEL];  // 8 KB
  __shared__ int   topi[4 * 16 * KSEL];  // 8 KB

  const int tid   = threadIdx.x;
  const int wave  = tid >> 5;
  const int lane  = tid & 31;
  const int batch = blockIdx.x / TILES;
  const int rtile = blockIdx.x % TILES;
  const int n0    = rtile * 16;
  const float* xb = xn + (size_t)batch * NPTS * LDIM;

  // stage A tile (rows n0..n0+15)
  for (int i = tid; i < 16 * LDIM; i += 128)
    As[i] = xb[(size_t)n0 * LDIM + i];
  // init per-wave top-k lists
  for (int i = lane; i < 16 * KSEL; i += 32) {
    topv[wave * 16 * KSEL + i] = -__builtin_inff();
    topi[wave * 16 * KSEL + i] = 0;
  }
  __syncthreads();

  // preload A-operand K-slices into registers (block-invariant across all B tiles)
  const int lm  = lane & 15;
  const int klo = (lane < 16) ? 0 : 2;
  v2f areg[32];
  #pragma unroll
  for (int s = 0; s < 32; ++s) {
    areg[s].x = As[lm * LDIM + 4 * s + klo];
    areg[s].y = As[lm * LDIM + 4 * s + klo + 1];
  }

  float* Bw   = Bs + wave * 16 * LDIM;
  float* tile = Bw;                       // first 256 floats reused for the D tile
  float* tv   = topv + (wave * 16 + lm) * KSEL;
  int*   ti   = topi + (wave * 16 + lm) * KSEL;
  float  mv   = -__builtin_inff();
  int    mp   = 0;

  for (int t = wave; t < TILES; t += 4) {
    const int m0 = t * 16;
    // stage B tile (rows m0..m0+15) into this wave's LDS region
    for (int i = lane; i < 16 * LDIM / 4; i += 32)
      ((float4*)Bw)[i] = ((const float4*)(xb + (size_t)m0 * LDIM))[i];

    // 16x16 tile of sim via 32 chained f32 WMMAs over K=128
    v8f acc = {0.f, 0.f, 0.f, 0.f, 0.f, 0.f, 0.f, 0.f};
    #pragma unroll
    for (int s = 0; s < 32; ++s) {
      v2f bb;
      bb.x = Bw[lm * LDIM + 4 * s + klo];
      bb.y = Bw[lm * LDIM + 4 * s + klo + 1];
      acc = __builtin_amdgcn_wmma_f32_16x16x4_f32(false, areg[s], false, bb,
                                                  (short)0, acc, false, false);
    }

    // scatter D (VGPR v: rows v / v+8; lanes: cols) into LDS tile
    const int rbase = (lane < 16) ? 0 : 8;
    #pragma unroll
    for (int v = 0; v < 8; ++v)
      tile[(rbase + v) * 16 + lm] = acc[v];

    // streaming top-k: lane r<16 owns row r; fast-path rejects v <= current min
    if (lane < 16) {
      for (int c = 0; c < 16; ++c) {
        float v = tile[lm * 16 + c];
        if (v > mv) {
          tv[mp] = v; ti[mp] = m0 + c;
          mv = tv[0]; mp = 0;
          for (int s2 = 1; s2 < KSEL; ++s2) {
            float t2 = tv[s2];
            if (t2 < mv) { mv = t2; mp = s2; }
          }
        }
      }
    }
  }
  __syncthreads();

  // merge the 4 per-wave lists -> exact top-32 per row (reuse As)
  float* mergedv = As;
  int*   mergedi = (int*)(As + 16 * KSEL);
  if (tid < 16) {
    const int r = tid;
    for (int j = 0; j < KSEL; ++j) {
      float best = -__builtin_inff(); int bpos = r * KSEL;
      for (int w = 0; w < 4; ++w)
        for (int s = 0; s < KSEL; ++s) {
          int p = (w * 16 + r) * KSEL + s;
          float v = topv[p];
          if (v > best) { best = v; bpos = p; }
        }
      mergedv[r * KSEL + j] = best;
      mergedi[r * KSEL + j] = topi[bpos];
      topv[bpos] = -__builtin_inff();
    }
  }
  __syncthreads();

  // out[b, n0+r, :] = relu(sum_j w_j * h[b, idx_j, :]); write both tuple halves
  const float* hb  = h + (size_t)batch * NPTS * HDIM;
  const size_t BNH = (size_t)BATCH * NPTS * HDIM;
  for (int e = tid; e < 16 * HDIM; e += 128) {
    int r = e >> 6, hc = e & 63;
    float acc2 = 0.f;
    #pragma unroll 8
    for (int j = 0; j < KSEL; ++j)
      acc2 += mergedv[r * KSEL + j] * hb[(size_t)mergedi[r * KSEL + j] * HDIM + hc];
    float o = fmaxf(acc2, 0.f);
    size_t oi = ((size_t)batch * NPTS + n0 + r) * HDIM + hc;
    out[oi]       = o;
    out[BNH + oi] = o;
  }
}

extern "C" void kernel_launch(void* const* d_in, const int* in_sizes, int n_in,
                              void* d_out, int out_size, void* d_ws, size_t ws_size,
                              hipStream_t stream) {
  const float* x = (const float*)d_in[0];
  const float* W = (const float*)d_in[1];
  const float* b = (const float*)d_in[2];
  // workspace: xn (2*10000*128 f32 = 10.24 MB) then h (2*10000*64 f32 = 5.12 MB)
  float* xn = (float*)d_ws;
  float* hh = xn + (size_t)BATCH * NPTS * LDIM;
  float* out = (float*)d_out;

  normalize_rows<<<(BATCH * NPTS + 7) / 8, 256, 0, stream>>>(x, xn);
  project_h<<<(BATCH * NPTS) / 4, 256, 0, stream>>>(x, W, b, hh);
  sim_topk_out<<<BATCH * TILES, 128, 0, stream>>>(xn, hh, out);
}